// ResidualAttentionBlock_49280454754675
// MI455X (gfx1250) — compile-verified
//
#include <hip/hip_runtime.h>
#include <hip/hip_bf16.h>
#include <math.h>

// ---------------- problem constants ----------------
#define BB 16
#define TT 81
#define DD 128
#define HH 8
#define MROWS (BB*TT)        // 1296 = 81 * 16, divisible by 16
#define NQKV  (HH*2*DD)      // 2048
#define TPAD  96             // 81 padded up to 6*16 for WMMA tiles
#define RSQRT_D 0.08838834764831845f  // 1/sqrt(128)

typedef __attribute__((ext_vector_type(2))) float v2f;
typedef __attribute__((ext_vector_type(8))) float v8f;

// D = A(16x4, f32) * B(4x16, f32) + C(16x16, f32)
// 8-arg pattern per probe: (neg_a, A, neg_b, B, c_mod, C, reuse_a, reuse_b)
__device__ __forceinline__ v8f wmma_f32_16x16x4(v2f a, v2f b, v8f c) {
  return __builtin_amdgcn_wmma_f32_16x16x4_f32(false, a, false, b, (short)0, c,
                                               false, false);
}

// ---------------------------------------------------------------------------
// Kernel A: y = x @ wqkv_w^T + b, split into q_ws (B,H,T,D) and
// v_ws (B,H,TPAD,D) (scaled by head_enabled). One wave = one 16x16 tile.
// A-frag: lane M = l%16, K = k0 + 2*(l/16) + {0,1}  -> contiguous float2
// B-frag: lane N = l%16 (row of W, since y = x W^T), same K -> float2
// ---------------------------------------------------------------------------
__global__ void qkv_gemm(const float* __restrict__ x, const float* __restrict__ w,
                         const float* __restrict__ bias,
                         const float* __restrict__ head_en,
                         float* __restrict__ q_ws, float* __restrict__ v_ws) {
  const int l    = threadIdx.x & 31;
  const int wave = threadIdx.x >> 5;
  const int m0 = blockIdx.x * 16;                    // 81 row tiles
  const int n0 = (blockIdx.y * 4 + wave) * 16;       // 128 col tiles
  const int lm = l & 15, lh = l >> 4;

  const float* arow = x + (size_t)(m0 + lm) * DD + 2 * lh;
  const float* brow = w + (size_t)(n0 + lm) * DD + 2 * lh;

  v8f c = {};
#pragma unroll 4
  for (int k0 = 0; k0 < DD; k0 += 4) {
    v2f a = *(const v2f*)(arow + k0);
    v2f b = *(const v2f*)(brow + k0);
    c = wmma_f32_16x16x4(a, b, c);
  }

  const int n  = n0 + lm;
  const int h  = n >> 8;         // n / (2*D)
  const int ci = n & 255;        // n % (2*D)
  const float bs = bias[n];
  const float he = head_en[h];
#pragma unroll
  for (int j = 0; j < 8; ++j) {
    const int m = m0 + j + 8 * lh;           // C/D layout: M = j + 8*(l/16)
    const int bidx = m / TT;
    const int t = m - bidx * TT;
    const float y = c[j] + bs;
    if (ci < DD) {
      q_ws[(((size_t)bidx * HH + h) * TT + t) * DD + ci] = y;
    } else {
      v_ws[(((size_t)bidx * HH + h) * TPAD + t) * DD + (ci - DD)] = y * he;
    }
  }
}

// ---------------------------------------------------------------------------
// Kernel B: L1-distance scores + softmax + mask. Block = (b,h,tq).
// k[b,tk,h,d] = x[b,tk,d] * wk[h,d] computed on the fly.
// Writes a_ws (B,H,TPAD,TPAD) and ap output for b==0.
// ---------------------------------------------------------------------------
__global__ void attn_scores(const float* __restrict__ x, const float* __restrict__ wk,
                            const float* __restrict__ msk,
                            const float* __restrict__ q_ws,
                            float* __restrict__ a_ws, float* __restrict__ ap_out) {
  const int tq = blockIdx.x, h = blockIdx.y, b = blockIdx.z;
  __shared__ float q_s[DD];
  __shared__ float wk_s[DD];
  __shared__ float sc[TT];

  const int tid = threadIdx.x;
  if (tid < DD) {
    q_s[tid]  = q_ws[(((size_t)b * HH + h) * TT + tq) * DD + tid];
    wk_s[tid] = wk[h * DD + tid];
  }
  __syncthreads();

  const int wave = tid >> 5, l = tid & 31;
  for (int tk = wave; tk < TT; tk += 8) {
    const float* xr = x + ((size_t)b * TT + tk) * DD;
    float s = 0.f;
#pragma unroll
    for (int i = 0; i < 4; ++i) {
      const int d = l + 32 * i;
      s += fabsf(q_s[d] - xr[d] * wk_s[d]);
    }
#pragma unroll
    for (int off = 16; off > 0; off >>= 1) s += __shfl_xor(s, off, 32);
    if (l == 0) sc[tk] = -s * RSQRT_D;
  }
  __syncthreads();

  if (wave == 0) {
    float mx = -1e30f;
    for (int tk = l; tk < TT; tk += 32) mx = fmaxf(mx, sc[tk]);
#pragma unroll
    for (int off = 16; off > 0; off >>= 1) mx = fmaxf(mx, __shfl_xor(mx, off, 32));
    float sum = 0.f;
    for (int tk = l; tk < TT; tk += 32) {
      const float e = __expf(sc[tk] - mx);
      sc[tk] = e;
      sum += e;
    }
#pragma unroll
    for (int off = 16; off > 0; off >>= 1) sum += __shfl_xor(sum, off, 32);
    const float inv = 1.0f / sum;
    for (int tk = l; tk < TT; tk += 32) {
      const float mv = msk[((size_t)h * TT + tq) * TT + tk];
      const float av = sc[tk] * inv * mv;
      a_ws[(((size_t)b * HH + h) * TPAD + tq) * TPAD + tk] = av;
      if (b == 0) ap_out[((size_t)tq * TT + tk) * HH + h] = av - 1.0f + mv;
    }
  }
}

// ---------------------------------------------------------------------------
// Kernel C: bo[b,t,d] = sum_h sum_s a[b,h,t,s] * v[b,h,s,d], fused with
// z = quick_gelu(bo+4) - 4. One wave per (b, t-tile, d-tile), accumulating
// over all 8 heads and K=TPAD (zero-padded rows contribute nothing).
// ---------------------------------------------------------------------------
__global__ void av_gemm(const float* __restrict__ a_ws, const float* __restrict__ v_ws,
                        float* __restrict__ z_ws) {
  const int l  = threadIdx.x & 31;
  const int b  = blockIdx.x;
  const int m0 = blockIdx.y * 16;   // tq tile, 0..5
  const int n0 = blockIdx.z * 16;   // d  tile, 0..7
  const int lm = l & 15, lh = l >> 4;

  v8f c = {};
  for (int h = 0; h < HH; ++h) {
    const float* abase = a_ws + (((size_t)b * HH + h) * TPAD + (m0 + lm)) * TPAD + 2 * lh;
    const float* vbase = v_ws + ((size_t)b * HH + h) * TPAD * DD + n0 + lm;
#pragma unroll 4
    for (int k0 = 0; k0 < TPAD; k0 += 4) {
      v2f a = *(const v2f*)(abase + k0);
      const int k = k0 + 2 * lh;
      v2f bf;
      bf[0] = vbase[(size_t)k * DD];
      bf[1] = vbase[(size_t)(k + 1) * DD];
      c = wmma_f32_16x16x4(a, bf, c);
    }
  }

  const int n = n0 + lm;
#pragma unroll
  for (int j = 0; j < 8; ++j) {
    const int t = m0 + j + 8 * lh;
    if (t < TT) {
      const float u = c[j] + 4.0f;
      const float g = u / (1.0f + __expf(-1.702f * u)) - 4.0f;  // quick_gelu(u)-4
      z_ws[((size_t)b * TT + t) * DD + n] = g;
    }
  }
}

// ---------------------------------------------------------------------------
// Kernel D: out = x + z @ fanout_w^T + fanout_b
// ---------------------------------------------------------------------------
__global__ void fanout_gemm(const float* __restrict__ z_ws, const float* __restrict__ wf,
                            const float* __restrict__ bfan, const float* __restrict__ x,
                            float* __restrict__ out) {
  const int l    = threadIdx.x & 31;
  const int wave = threadIdx.x >> 5;
  const int m0 = blockIdx.x * 16;                 // 81 row tiles
  const int n0 = (blockIdx.y * 4 + wave) * 16;    // 8 col tiles
  const int lm = l & 15, lh = l >> 4;

  const float* arow = z_ws + (size_t)(m0 + lm) * DD + 2 * lh;
  const float* brow = wf   + (size_t)(n0 + lm) * DD + 2 * lh;

  v8f c = {};
#pragma unroll 4
  for (int k0 = 0; k0 < DD; k0 += 4) {
    v2f a = *(const v2f*)(arow + k0);
    v2f b = *(const v2f*)(brow + k0);
    c = wmma_f32_16x16x4(a, b, c);
  }

  const int n = n0 + lm;
  const float bs = bfan[n];
#pragma unroll
  for (int j = 0; j < 8; ++j) {
    const int m = m0 + j + 8 * lh;
    out[(size_t)m * DD + n] = x[(size_t)m * DD + n] + c[j] + bs;
  }
}

// ---------------------------------------------------------------------------
extern "C" void kernel_launch(void* const* d_in, const int* in_sizes, int n_in,
                              void* d_out, int out_size, void* d_ws, size_t ws_size,
                              hipStream_t stream) {
  (void)in_sizes; (void)n_in; (void)out_size; (void)ws_size;

  const float* x    = (const float*)d_in[0];
  const float* msk  = (const float*)d_in[1];
  const float* wqkv = (const float*)d_in[2];
  const float* bqkv = (const float*)d_in[3];
  const float* wk   = (const float*)d_in[4];
  const float* wf   = (const float*)d_in[5];
  const float* bfan = (const float*)d_in[6];
  const float* hen  = (const float*)d_in[7];

  float* out_main = (float*)d_out;                 // (B,T,D) = 165888 floats
  float* out_ap   = out_main + (size_t)BB*TT*DD;   // (T,T,H) =  52488 floats

  // workspace layout (floats)
  float* q_ws = (float*)d_ws;                                  // B*H*T*D
  float* v_ws = q_ws + (size_t)BB*HH*TT*DD;                    // B*H*TPAD*D
  float* a_ws = v_ws + (size_t)BB*HH*TPAD*DD;                  // B*H*TPAD*TPAD
  float* z_ws = a_ws + (size_t)BB*HH*TPAD*TPAD;                // MROWS*D

  // zero the padded regions so WMMA K/M padding contributes nothing
  hipMemsetAsync(v_ws, 0, (size_t)BB*HH*TPAD*DD   * sizeof(float), stream);
  hipMemsetAsync(a_ws, 0, (size_t)BB*HH*TPAD*TPAD * sizeof(float), stream);

  // 1) QKV projection (WMMA f32), split q/v, bias, head_enabled
  qkv_gemm<<<dim3(MROWS/16, NQKV/(16*4)), 128, 0, stream>>>(x, wqkv, bqkv, hen,
                                                            q_ws, v_ws);
  // 2) L1 attention scores + softmax + mask (+ ap side output)
  attn_scores<<<dim3(TT, HH, BB), 256, 0, stream>>>(x, wk, msk, q_ws, a_ws, out_ap);
  // 3) a @ v summed over heads, fused quick-GELU shift (WMMA f32)
  av_gemm<<<dim3(BB, TPAD/16, DD/16), 32, 0, stream>>>(a_ws, v_ws, z_ws);
  // 4) fanout GEMM + bias + residual (WMMA f32)
  fanout_gemm<<<dim3(MROWS/16, (DD/16)/4), 128, 0, stream>>>(z_ws, wf, bfan, x,
                                                             out_main);
}